// FFMLayer_67473936220476
// MI455X (gfx1250) — compile-verified
//
#include <hip/hip_runtime.h>

#define B_ROWS   131072
#define N_FEAT   128
#define N_FIELD  16
#define FACTOR   16
#define WAVES_PER_BLOCK 4
#define XS_STRIDE 132   // 128 + 4 pad: row stride 4 banks -> conflict-free gathers

typedef __attribute__((ext_vector_type(2))) float v2f;
typedef __attribute__((ext_vector_type(8))) float v8f;

// ---------------------------------------------------------------------------
// Kernel 1: build the fixed quadratic-form matrix, stored directly in WMMA
// B-operand lane order.
//   logical A[i][j] = (i==j) ? 0 : 0.5 * dot(w[i, field[j], :], w[j, field[i], :])
// Swizzled layout: for lane l (h=l>>4, n=l&15), k-pair kk, column block t,
// the float4 at ((t*16+kk)*32 + l)*4 holds rows {8kk+2h, +1, 8kk+4+2h, +1}
// of column 16t+n -> main kernel reads one coalesced b128 per two WMMAs.
// ---------------------------------------------------------------------------
__global__ void ffm_build_A(const float* __restrict__ w,
                            const int* __restrict__ field,
                            float* __restrict__ A) {
  int idx = blockIdx.x * blockDim.x + threadIdx.x;
  if (idx >= N_FEAT * N_FEAT) return;
  int i = idx >> 7;              // logical row
  int j = idx & (N_FEAT - 1);    // logical col
  const float* wi = w + (i * N_FIELD + field[j]) * FACTOR;
  const float* wj = w + (j * N_FIELD + field[i]) * FACTOR;
  float s = 0.f;
#pragma unroll
  for (int d = 0; d < FACTOR; ++d) s += wi[d] * wj[d];
  const float val = (i == j) ? 0.f : 0.5f * s;

  // scatter into B-operand swizzled position
  const int t  = j >> 4,  n = j & 15;
  const int kk = i >> 3,  r = i & 7;
  const int h  = (r >> 1) & 1;               // half-wave
  const int q  = ((r >> 2) << 1) | (r & 1);  // component within float4
  const int lane = h * 16 + n;
  A[(((t * 16 + kk) * 32) + lane) * 4 + q] = val;
}

// ---------------------------------------------------------------------------
// Kernel 2: out[b] = x_b^T A x_b, fused GEMM + row-dot via V_WMMA_F32_16X16X4_F32
// One wave32 per 16 batch rows. X tile staged in LDS; swizzled A streamed from
// L2/WGP$ with fully-coalesced b128 loads (one per two WMMAs).
// ---------------------------------------------------------------------------
__global__ __launch_bounds__(WAVES_PER_BLOCK * 32)
void ffm_quadform(const float* __restrict__ x,
                  const float* __restrict__ A,
                  float* __restrict__ out) {
  __shared__ float xs[WAVES_PER_BLOCK][16][XS_STRIDE];

  const int wave = threadIdx.x >> 5;
  const int lane = threadIdx.x & 31;
  const int half = lane >> 4;      // 0: lanes 0-15, 1: lanes 16-31
  const int l16  = lane & 15;
  const long tile = (long)blockIdx.x * WAVES_PER_BLOCK + wave; // 16-row tile
  const float* xt = x + tile * 16 * N_FEAT;

  // Stage 16x128 fp32 X tile into LDS (coalesced b128 loads per row).
#pragma unroll
  for (int r = 0; r < 16; ++r) {
    const float4 v = *(const float4*)(xt + r * N_FEAT + lane * 4);
    xs[wave][r][lane * 4 + 0] = v.x;
    xs[wave][r][lane * 4 + 1] = v.y;
    xs[wave][r][lane * 4 + 2] = v.z;
    xs[wave][r][lane * 4 + 3] = v.w;
  }
  // Wave-private tile: RAW on LDS resolved by compiler-inserted s_wait_dscnt.

  float rowacc[8];
#pragma unroll
  for (int v = 0; v < 8; ++v) rowacc[v] = 0.f;

  const float4* __restrict__ Ab = (const float4*)A;

  for (int t = 0; t < 8; ++t) {              // output column blocks of 16
    v8f c = {};                              // C/D: 16x16 f32 in 8 VGPRs
    const float4* bp = Ab + (size_t)t * 16 * 32 + lane;
#pragma unroll
    for (int kk = 0; kk < 16; ++kk) {        // K = 128, two k-steps per iter
      const float4 bv = bp[kk * 32];         // coalesced b128: B for k=2kk,2kk+1
      // A-operand (16x4 f32): VGPR0 = K (lanes 0-15) / K+2 (lanes 16-31),
      // VGPR1 = K+1 / K+3. Lane reads its row's contiguous float2 from LDS.
      const float2 a0 = *(const float2*)(&xs[wave][l16][8 * kk + 2 * half]);
      const float2 a1 = *(const float2*)(&xs[wave][l16][8 * kk + 4 + 2 * half]);

      v2f av0; av0.x = a0.x; av0.y = a0.y;
      v2f bv0; bv0.x = bv.x; bv0.y = bv.y;
      c = __builtin_amdgcn_wmma_f32_16x16x4_f32(
              false, av0, false, bv0, (short)0, c, false, false);

      v2f av1; av1.x = a1.x; av1.y = a1.y;
      v2f bv1; bv1.x = bv.z; bv1.y = bv.w;
      c = __builtin_amdgcn_wmma_f32_16x16x4_f32(
              false, av1, false, bv1, (short)0, c, false, false);
    }
    // Fold Y tile into per-row dot: D layout lane=N, vgpr v -> M = v + 8*half.
    const int col = t * 16 + l16;
#pragma unroll
    for (int v = 0; v < 8; ++v)
      rowacc[v] += c[v] * xs[wave][v + 8 * half][col];
  }

  // Reduce the 16 lanes of each half-wave (xor masks < 16 stay within half).
#pragma unroll
  for (int v = 0; v < 8; ++v) {
    float s = rowacc[v];
    s += __shfl_xor(s, 1, 32);
    s += __shfl_xor(s, 2, 32);
    s += __shfl_xor(s, 4, 32);
    s += __shfl_xor(s, 8, 32);
    if (l16 == 0) out[tile * 16 + v + 8 * half] = s;  // lanes 0 and 16 store
  }
}

// ---------------------------------------------------------------------------
extern "C" void kernel_launch(void* const* d_in, const int* in_sizes, int n_in,
                              void* d_out, int out_size, void* d_ws, size_t ws_size,
                              hipStream_t stream) {
  const float* x     = (const float*)d_in[0];   // (B, 128) fp32
  const float* w     = (const float*)d_in[1];   // (128, 16, 16) fp32
  const int*   field = (const int*)d_in[2];     // (128,) int32
  float* out = (float*)d_out;                   // (B, 1) fp32
  float* A   = (float*)d_ws;                    // 128*128 fp32 = 64 KB scratch

  ffm_build_A<<<(N_FEAT * N_FEAT + 255) / 256, 256, 0, stream>>>(w, field, A);

  const int tiles  = B_ROWS / 16;                       // 8192
  const int blocks = tiles / WAVES_PER_BLOCK;           // 2048
  ffm_quadform<<<blocks, WAVES_PER_BLOCK * 32, 0, stream>>>(x, A, out);
}